// MeanAggregator_91285234909620
// MI455X (gfx1250) — compile-verified
//
#include <hip/hip_runtime.h>

typedef __attribute__((ext_vector_type(2))) float v2f;
typedef __attribute__((ext_vector_type(8))) float v8f;

namespace {
constexpr int kS = 32;              // sampled neighbors per node
constexpr int kD = 128;             // embedding dim
constexpr int kWavesPerBlock = 8;   // wave32 -> 256 threads
constexpr int kBlockThreads = kWavesPerBlock * 32;
}

// Mean aggregation as a ones-matrix WMMA reduction:
//   out[node][d] = (1/32) * sum_s weight[idx[node][s]][d]
// One wave32 per node. Per 4-neighbor chunk: 8 independent
// V_WMMA_F32_16X16X4_F32 ops (one per 16-dim group) with A == ones, so
// D[m][n] = sum_k B[k][n]; C-accumulation over 8 chunks gives the full
// 32-neighbor sum per column. A == ones makes the result invariant to the
// hardware's K <-> (B-VGPR, lane-half) mapping; we only rely on N == lane%16.
//
// B tiles are double-buffered across chunks and the schedule is pinned with
// sched_barrier so each chunk's 16 loads are issued while the previous
// chunk's WMMAs execute (16-32 outstanding loads/wave instead of 2).
__global__ __launch_bounds__(kBlockThreads)
void mean_agg_wmma_kernel(const int* __restrict__ neigh_idx,
                          const float* __restrict__ weight,
                          float* __restrict__ out,
                          int n_nodes)
{
    const int lane = threadIdx.x & 31;
    const int node = blockIdx.x * kWavesPerBlock + (threadIdx.x >> 5);
    if (node >= n_nodes) return;     // wave-uniform: EXEC stays all-ones

    const int col  = lane & 15;      // output column (N) this lane feeds
    const int half = lane >> 4;      // which 16-lane half

    // One neighbor index per lane; broadcast via ds_bpermute.
    const int myidx = neigh_idx[node * kS + lane];

    v8f acc[8] = {};                 // 8 dim-groups x (16x16 f32 tile)
    const v2f ones = {1.0f, 1.0f};   // A matrix: all ones (layout-agnostic)

    v2f b[2][8];                     // double-buffered B tiles

    // ---- prologue: addresses + loads for chunk 0 ----
    {
        const int i0 = __builtin_amdgcn_ds_bpermute((0 +     half) << 2, myidx);
        const int i1 = __builtin_amdgcn_ds_bpermute((0 + 2 + half) << 2, myidx);
        const float* __restrict__ p0 = weight + i0 * kD + col;
        const float* __restrict__ p1 = weight + i1 * kD + col;
        #pragma unroll
        for (int g = 0; g < 8; ++g) {
            b[0][g].x = p0[g * 16];  // 64B coalesced per half-wave
            b[0][g].y = p1[g * 16];
        }
    }

    #pragma unroll
    for (int c = 0; c < 8; ++c) {
        const int cur = c & 1;
        const int nxt = cur ^ 1;

        // issue next chunk's 16 loads before consuming this chunk
        if (c < 7) {
            const int s = 4 * (c + 1);
            const int i0 = __builtin_amdgcn_ds_bpermute((s +     half) << 2, myidx);
            const int i1 = __builtin_amdgcn_ds_bpermute((s + 2 + half) << 2, myidx);
            const float* __restrict__ p0 = weight + i0 * kD + col;
            const float* __restrict__ p1 = weight + i1 * kD + col;
            #pragma unroll
            for (int g = 0; g < 8; ++g) {
                b[nxt][g].x = p0[g * 16];
                b[nxt][g].y = p1[g * 16];
            }
        }

        // Keep the prefetch loads above the WMMA block: no reordering across.
        __builtin_amdgcn_sched_barrier(0);

        #pragma unroll
        for (int g = 0; g < 8; ++g) {
            acc[g] = __builtin_amdgcn_wmma_f32_16x16x4_f32(
                /*neg_a=*/false, ones,
                /*neg_b=*/false, b[cur][g],
                /*c_mod=*/(short)0, acc[g],
                /*reuse_a=*/false, /*reuse_b=*/false);
        }

        __builtin_amdgcn_sched_barrier(0);
    }

    // D layout: VGPR0, lanes 0-15 -> (M=0, N=lane); every column is the
    // 32-neighbor sum for dim g*16+col. Lanes 16-31 hold duplicates (M=8),
    // so only the low half stores.
    if (half == 0) {
        const float inv = 1.0f / 32.0f;
        #pragma unroll
        for (int g = 0; g < 8; ++g) {
            out[node * kD + g * 16 + col] = acc[g][0] * inv;
        }
    }
}

extern "C" void kernel_launch(void* const* d_in, const int* in_sizes, int n_in,
                              void* d_out, int out_size, void* d_ws, size_t ws_size,
                              hipStream_t stream) {
    const int*   neigh_idx = (const int*)d_in[0];   // [N, 32] int32
    const float* weight    = (const float*)d_in[1]; // [M, 128] fp32
    float*       out       = (float*)d_out;         // [N, 128] fp32

    const int n_nodes = in_sizes[0] / kS;
    const int blocks  = (n_nodes + kWavesPerBlock - 1) / kWavesPerBlock;
    mean_agg_wmma_kernel<<<blocks, kBlockThreads, 0, stream>>>(
        neigh_idx, weight, out, n_nodes);
}